// GraphGAT_3418793967714
// MI455X (gfx1250) — compile-verified
//
#include <hip/hip_runtime.h>

#define NNODES 50000
#define NEDGES 800000
#define ETOT   (NEDGES + NNODES)      // 850000 with self loops
#define INDIM  128
#define HID1   16
#define HEADS1 8
#define F1     (HEADS1 * HID1)        // 128
#define OUTD   64
#define ENC_NEG_INF ((int)0x807FFFFF) // enc(-inf) under order-preserving map

typedef __attribute__((ext_vector_type(2))) float v2f;
typedef __attribute__((ext_vector_type(8))) float v8f;

// ---- order-preserving float<->int map so float segment-max is one atomicMax ----
static __device__ __forceinline__ int enc_ord(float f) {
  int i = __float_as_int(f);
  return i >= 0 ? i : (i ^ 0x7FFFFFFF);
}
static __device__ __forceinline__ float dec_ord(int i) {
  return __int_as_float(i >= 0 ? i : (i ^ 0x7FFFFFFF));
}

// ============================================================================
// C[M x NC] = X[M x K] * W[K x NC], fp32, V_WMMA_F32_16X16X4_F32.
// One wave -> 16 rows x NC cols (NT=NC/16 accumulators), A-frag reused across
// column tiles. M must be a multiple of 16 (50000 = 3125*16).
// A 16x4 frag: lane<16 -> M=lane, K={k,k+1}; lane>=16 -> M=lane-16, K={k+2,k+3}
// B 4x16 frag: mirrored (rows across half-waves, cols across lanes 0..15)
// D: vgpr r -> row r (+8 for upper half-wave), col = lane&15
// ============================================================================
template <int NC>
__global__ void gemm_wmma_f32(const float* __restrict__ X, const float* __restrict__ W,
                              float* __restrict__ C, int M, int K) {
  constexpr int NT = NC / 16;
  const int wave = (blockIdx.x * blockDim.x + threadIdx.x) >> 5;
  const int lane = threadIdx.x & 31;
  const int row0 = wave * 16;
  if (row0 >= M) return;                       // wave-uniform: EXEC stays all-1s
  const int mrow  = row0 + (lane & 15);
  const int khalf = (lane >> 4) << 1;          // 0 or 2
  const int ncol  = lane & 15;

  v8f zero = {};
  v8f acc[NT];
#pragma unroll
  for (int t = 0; t < NT; ++t) acc[t] = zero;

  const float* xrow = X + (size_t)mrow * K;
  for (int k = 0; k < K; k += 4) {
    const int ka = k + khalf;
    v2f a = *(const v2f*)(xrow + ka);          // contiguous pair, 8B aligned
#pragma unroll
    for (int t = 0; t < NT; ++t) {
      const int n = t * 16 + ncol;
      v2f b;
      b.x = W[(size_t)ka * NC + n];
      b.y = W[(size_t)(ka + 1) * NC + n];
      acc[t] = __builtin_amdgcn_wmma_f32_16x16x4_f32(
          false, a, false, b, (short)0, acc[t], false, false);
    }
  }
  const int rbase = row0 + ((lane >> 4) << 3);
#pragma unroll
  for (int t = 0; t < NT; ++t)
#pragma unroll
    for (int r = 0; r < 8; ++r)
      C[(size_t)(rbase + r) * NC + t * 16 + ncol] = acc[t][r];
}

// a_src[n,h] = <h[n,h,:], att_src[h,:]>  (and same for dst)
__global__ void att_scores_k(const float* __restrict__ h, const float* __restrict__ as,
                             const float* __restrict__ ad, float* __restrict__ osrc,
                             float* __restrict__ odst, int hshift, int Cc, int total) {
  int idx = blockIdx.x * blockDim.x + threadIdx.x;
  if (idx >= total) return;
  int H = 1 << hshift;
  int hd = idx & (H - 1);
  const float* hp = h + (size_t)idx * Cc;      // idx == n*H+hd
  const float* ps = as + hd * Cc;
  const float* pd = ad + hd * Cc;
  float ss = 0.f, sd = 0.f;
  for (int c = 0; c < Cc; ++c) { float v = hp[c]; ss += v * ps[c]; sd += v * pd[c]; }
  osrc[idx] = ss;
  odst[idx] = sd;
}

__global__ void fill_i32_k(int* __restrict__ p, int v, int n) {
  int i = blockIdx.x * blockDim.x + threadIdx.x;
  if (i < n) p[i] = v;
}

__global__ void bcast_bias_k(float* __restrict__ o, const float* __restrict__ b,
                             int fmask, int total) {
  int i = blockIdx.x * blockDim.x + threadIdx.x;
  if (i < total) o[i] = b[i & fmask];
}

__global__ void elu_k(const float* __restrict__ in, float* __restrict__ o, int total) {
  int i = blockIdx.x * blockDim.x + threadIdx.x;
  if (i >= total) return;
  float v = in[i];
  o[i] = v > 0.f ? v : (__expf(v) - 1.f);
}

// pass A: e = leaky_relu(a_src[src]+a_dst[dst]); stash e; segment-max into mb
__global__ void edge_logits_max_k(const int* __restrict__ ei, int E, int hshift,
                                  const float* __restrict__ asrc,
                                  const float* __restrict__ adst,
                                  float* __restrict__ eb, int* __restrict__ mb,
                                  int total) {
  int tid = blockIdx.x * blockDim.x + threadIdx.x;
  if (tid >= total) return;
  int H = 1 << hshift;
  int e = tid >> hshift, hd = tid & (H - 1);
  int src, dst;
  if (e < E) { src = ei[e]; dst = ei[E + e]; } else { src = dst = e - E; }
  float v = asrc[(src << hshift) + hd] + adst[(dst << hshift) + hd];
  v = v > 0.f ? v : 0.2f * v;                  // leaky_relu slope 0.2
  eb[tid] = v;
  atomicMax(mb + (dst << hshift) + hd, enc_ord(v));
}

// pass B: p = exp(e - m[dst]); stash p; segment-sum into sb
__global__ void edge_expsum_k(const int* __restrict__ ei, int E, int hshift,
                              const int* __restrict__ mb, float* __restrict__ eb,
                              float* __restrict__ sb, int total) {
  int tid = blockIdx.x * blockDim.x + threadIdx.x;
  if (tid >= total) return;
  int H = 1 << hshift;
  int e = tid >> hshift, hd = tid & (H - 1);
  int dst = (e < E) ? ei[E + e] : e - E;
  float p = __expf(eb[tid] - dec_ord(mb[(dst << hshift) + hd]));
  eb[tid] = p;
  atomicAdd(sb + (dst << hshift) + hd, p);
}

// pass C: out[dst,f] += (p / s[dst]) * h[src,f]   (thread per edge-feature)
__global__ void edge_aggregate_k(const int* __restrict__ ei, int E, int fshift, int cshift,
                                 const float* __restrict__ eb, const float* __restrict__ sb,
                                 const float* __restrict__ hsrc, float* __restrict__ outp,
                                 int total) {
  int tid = blockIdx.x * blockDim.x + threadIdx.x;
  if (tid >= total) return;
  int F = 1 << fshift;
  int e = tid >> fshift, f = tid & (F - 1);
  int hd = f >> cshift;
  int hshift = fshift - cshift;                // H = F / C
  int src, dst;
  if (e < E) { src = ei[e]; dst = ei[E + e]; } else { src = dst = e - E; }
  float alpha = eb[(e << hshift) + hd] / (sb[(dst << hshift) + hd] + 1e-16f);
  atomicAdd(outp + ((size_t)dst << fshift) + f,
            alpha * hsrc[((size_t)src << fshift) + f]);
}

static inline int cdiv(long long a, int b) { return (int)((a + b - 1) / b); }

extern "C" void kernel_launch(void* const* d_in, const int* in_sizes, int n_in,
                              void* d_out, int out_size, void* d_ws, size_t ws_size,
                              hipStream_t stream) {
  const float* x        = (const float*)d_in[0];
  const int*   ei       = (const int*)d_in[1];   // [2,E]: src row then dst row
  const float* W1       = (const float*)d_in[2];
  const float* att_src1 = (const float*)d_in[3];
  const float* att_dst1 = (const float*)d_in[4];
  const float* b1       = (const float*)d_in[5];
  const float* W2       = (const float*)d_in[6];
  const float* att_src2 = (const float*)d_in[7];
  const float* att_dst2 = (const float*)d_in[8];
  const float* b2       = (const float*)d_in[9];
  float* out = (float*)d_out;

  // workspace carve-up (~98 MB)
  float* h1   = (float*)d_ws;                         // N*128
  float* out1 = h1   + (size_t)NNODES * F1;           // N*128
  float* h2   = out1 + (size_t)NNODES * F1;           // N*64
  float* a_s  = h2   + (size_t)NNODES * OUTD;         // N*8 (layer2 reuses N)
  float* a_d  = a_s  + (size_t)NNODES * HEADS1;       // N*8
  int*   mb   = (int*)(a_d + (size_t)NNODES * HEADS1);// N*8
  float* sb   = (float*)(mb + (size_t)NNODES * HEADS1);// N*8
  float* eb   = sb + (size_t)NNODES * HEADS1;         // ET*8

  const int B = 256;
  const int gemm_threads = (NNODES / 16) * 32;        // 3125 waves

  // ---------------- Layer 1 (H=8, C=16, F=128) ----------------
  gemm_wmma_f32<F1><<<cdiv(gemm_threads, B), B, 0, stream>>>(x, W1, h1, NNODES, INDIM);
  att_scores_k<<<cdiv((long long)NNODES * HEADS1, B), B, 0, stream>>>(
      h1, att_src1, att_dst1, a_s, a_d, 3, HID1, NNODES * HEADS1);
  fill_i32_k<<<cdiv((long long)NNODES * HEADS1, B), B, 0, stream>>>(mb, ENC_NEG_INF, NNODES * HEADS1);
  fill_i32_k<<<cdiv((long long)NNODES * HEADS1, B), B, 0, stream>>>((int*)sb, 0, NNODES * HEADS1);
  bcast_bias_k<<<cdiv((long long)NNODES * F1, B), B, 0, stream>>>(out1, b1, F1 - 1, NNODES * F1);
  edge_logits_max_k<<<cdiv((long long)ETOT * HEADS1, B), B, 0, stream>>>(
      ei, NEDGES, 3, a_s, a_d, eb, mb, ETOT * HEADS1);
  edge_expsum_k<<<cdiv((long long)ETOT * HEADS1, B), B, 0, stream>>>(
      ei, NEDGES, 3, mb, eb, sb, ETOT * HEADS1);
  edge_aggregate_k<<<cdiv((long long)ETOT * F1, B), B, 0, stream>>>(
      ei, NEDGES, 7, 4, eb, sb, h1, out1, ETOT * F1);
  elu_k<<<cdiv((long long)NNODES * F1, B), B, 0, stream>>>(out1, h1, NNODES * F1); // z -> h1

  // ---------------- Layer 2 (H=1, C=64, F=64) ----------------
  gemm_wmma_f32<OUTD><<<cdiv(gemm_threads, B), B, 0, stream>>>(h1, W2, h2, NNODES, F1);
  att_scores_k<<<cdiv((long long)NNODES, B), B, 0, stream>>>(
      h2, att_src2, att_dst2, a_s, a_d, 0, OUTD, NNODES);
  fill_i32_k<<<cdiv((long long)NNODES, B), B, 0, stream>>>(mb, ENC_NEG_INF, NNODES);
  fill_i32_k<<<cdiv((long long)NNODES, B), B, 0, stream>>>((int*)sb, 0, NNODES);
  bcast_bias_k<<<cdiv((long long)NNODES * OUTD, B), B, 0, stream>>>(out, b2, OUTD - 1, NNODES * OUTD);
  edge_logits_max_k<<<cdiv((long long)ETOT, B), B, 0, stream>>>(
      ei, NEDGES, 0, a_s, a_d, eb, mb, ETOT);
  edge_expsum_k<<<cdiv((long long)ETOT, B), B, 0, stream>>>(
      ei, NEDGES, 0, mb, eb, sb, ETOT);
  edge_aggregate_k<<<cdiv((long long)ETOT * OUTD, B), B, 0, stream>>>(
      ei, NEDGES, 6, 6, eb, sb, h2, out, ETOT * OUTD);

  (void)in_sizes; (void)n_in; (void)out_size; (void)ws_size;
}